// Linear_34308198760561
// MI455X (gfx1250) — compile-verified
//
#include <hip/hip_runtime.h>

// Problem constants (from reference)
#define BATCHN 64
#define NT     4096
#define NXD    128
#define NUD    32
#define NYD    32
// Chunked-scan parameters: 128 chunks x 32 steps = 4096
#define NCHUNK 128
#define CLEN   32
// LDS strides (floats)
#define STX    132   // X state rows (A-operand, column access): 132%64=4 -> conflict-free
#define STB    160   // B-operand matrices (row access): 160%64=32 -> half-waves hit disjoint banks

typedef __attribute__((ext_vector_type(2))) float v2f;
typedef __attribute__((ext_vector_type(8))) float v8f;

static __device__ __forceinline__ v8f wmma4(v2f a, v2f b, v8f c) {
  // D = A(16x4) * B(4x16) + C(16x16), fp32
  return __builtin_amdgcn_wmma_f32_16x16x4_f32(false, a, false, b, (short)0, c, false, false);
}

// A-operand (M=16,K=4): lanes 0-15 -> M rows, VGPR0/1 = K0/K1; lanes 16-31 -> K2/K3
static __device__ __forceinline__ v2f ldA(const float* M, int stride, int mbase, int kk, int lane) {
  int m = mbase + (lane & 15);
  int k = kk + ((lane >> 4) << 1);
  v2f a;
  a.x = M[m * stride + k];
  a.y = M[m * stride + k + 1];
  return a;
}

// B-operand (K=4,N=16): VGPR0 = rows K0(l0-15)/K1(l16-31), VGPR1 = rows K2/K3
static __device__ __forceinline__ v2f ldB(const float* M, int stride, int kk, int nbase, int lane) {
  int n = nbase + (lane & 15);
  int k = kk + (lane >> 4);
  v2f b;
  b.x = M[k * stride + n];
  b.y = M[(k + 2) * stride + n];
  return b;
}

// C/D 16x16 fp32 accumulator layout: VGPR r -> M = mbase + r + 8*(lane>=16), N = nbase + (lane&15)
static __device__ __forceinline__ void stAccLds(float* M, int stride, int mbase, int nbase, int lane, v8f acc) {
  int n = nbase + (lane & 15);
  int mo = mbase + ((lane >> 4) << 3);
#pragma unroll
  for (int r = 0; r < 8; ++r) M[(mo + r) * stride + n] = acc[r];
}

// ---------------------------------------------------------------------------
// Kernel 1: Apow = A^32 via 5 WMMA squarings (single workgroup)
// ---------------------------------------------------------------------------
__global__ __launch_bounds__(256) void lti_power_kernel(const float* __restrict__ A,
                                                        float* __restrict__ wsApow) {
  __shared__ float sP[NXD * STX];
  __shared__ float sQ[NXD * STX];
  const int tid = threadIdx.x, lane = tid & 31, wave = tid >> 5;

  for (int i = tid; i < NXD * NXD; i += 256) {
    int r = i >> 7, c = i & (NXD - 1);
    sP[r * STX + c] = A[i];
  }
  __syncthreads();

  float* pa = sP;
  float* pb = sQ;
  for (int it = 0; it < 5; ++it) {   // A^2, A^4, A^8, A^16, A^32
    v8f acc[8];
#pragma unroll
    for (int j = 0; j < 8; ++j) { v8f z = {}; acc[j] = z; }
    for (int kk = 0; kk < NXD; kk += 4) {
      v2f a = ldA(pa, STX, wave * 16, kk, lane);
#pragma unroll
      for (int ct = 0; ct < 8; ++ct) {
        v2f b = ldB(pa, STX, kk, ct * 16, lane);
        acc[ct] = wmma4(a, b, acc[ct]);
      }
    }
    __syncthreads();
#pragma unroll
    for (int ct = 0; ct < 8; ++ct) stAccLds(pb, STX, wave * 16, ct * 16, lane, acc[ct]);
    __syncthreads();
    float* t = pa; pa = pb; pb = t;
  }
  for (int i = tid; i < NXD * NXD; i += 256) {
    int r = i >> 7, c = i & (NXD - 1);
    wsApow[i] = pa[r * STX + c];
  }
}

// ---------------------------------------------------------------------------
// Kernel 2: per-chunk zero-state contribution c_j = final state of chunk j
//           starting from x = 0  (grid = NCHUNK blocks)
// ---------------------------------------------------------------------------
__global__ __launch_bounds__(256) void lti_phase1_kernel(const float* __restrict__ d,
                                                         const float* __restrict__ A,
                                                         const float* __restrict__ B,
                                                         float* __restrict__ wsC) {
  __shared__ float sAT[NXD * STB];   // AT[k][n] = A[n][k]
  __shared__ float sBT[NUD * STB];   // BT[k][n] = B[n][k]
  __shared__ float sX[BATCHN * STX];
  const int tid = threadIdx.x, lane = tid & 31, wave = tid >> 5;
  const int chunk = blockIdx.x;

  for (int i = tid; i < NXD * NXD; i += 256) {
    int n = i >> 7, k = i & (NXD - 1);
    sAT[k * STB + n] = A[i];
  }
  for (int i = tid; i < NXD * NUD; i += 256) {
    int n = i >> 5, k = i & (NUD - 1);
    sBT[k * STB + n] = B[i];
  }
  for (int i = tid; i < BATCHN * STX; i += 256) sX[i] = 0.0f;
  __syncthreads();

  const int mbase = (wave >> 1) * 16;
  const int cb = (wave & 1) * 4;      // 4 column tiles of 16 -> 64 nx columns
  const int t0 = chunk * CLEN;

  for (int s = 0; s < CLEN; ++s) {
    const int t = t0 + s;
    v8f acc[4];
#pragma unroll
    for (int j = 0; j < 4; ++j) { v8f z = {}; acc[j] = z; }
    // X @ A^T
    for (int kk = 0; kk < NXD; kk += 4) {
      v2f a = ldA(sX, STX, mbase, kk, lane);
#pragma unroll
      for (int j = 0; j < 4; ++j)
        acc[j] = wmma4(a, ldB(sAT, STB, kk, (cb + j) * 16, lane), acc[j]);
    }
    // U @ B^T  (U streamed from global: d[b][t][k])
    for (int kk = 0; kk < NUD; kk += 4) {
      int m = mbase + (lane & 15);
      int k = kk + ((lane >> 4) << 1);
      const float* p = d + ((size_t)m * NT + t) * NUD + k;
      v2f a; a.x = p[0]; a.y = p[1];
#pragma unroll
      for (int j = 0; j < 4; ++j)
        acc[j] = wmma4(a, ldB(sBT, STB, kk, (cb + j) * 16, lane), acc[j]);
    }
    __syncthreads();
#pragma unroll
    for (int j = 0; j < 4; ++j) stAccLds(sX, STX, mbase, (cb + j) * 16, lane, acc[j]);
    __syncthreads();
  }

  float* cdst = wsC + (size_t)chunk * (BATCHN * NXD);
  for (int q = tid; q < (BATCHN * NXD) / 4; q += 256) {
    int b = q >> 5, i = (q & 31) << 2;
    float4 v = *(const float4*)&sX[b * STX + i];
    *(float4*)&cdst[b * NXD + i] = v;
  }
}

// ---------------------------------------------------------------------------
// Kernel 3: serial combine  s_0 = 0;  s_{j+1} = A^32 s_j + c_j  (1 workgroup)
// ---------------------------------------------------------------------------
__global__ __launch_bounds__(256) void lti_combine_kernel(const float* __restrict__ wsApow,
                                                          const float* __restrict__ wsC,
                                                          float* __restrict__ wsS) {
  __shared__ float sAT[NXD * STB];   // (A^32)^T
  __shared__ float sX[BATCHN * STX];
  const int tid = threadIdx.x, lane = tid & 31, wave = tid >> 5;

  for (int i = tid; i < NXD * NXD; i += 256) {
    int n = i >> 7, k = i & (NXD - 1);
    sAT[k * STB + n] = wsApow[i];
  }
  for (int i = tid; i < BATCHN * STX; i += 256) sX[i] = 0.0f;
  __syncthreads();

  const int mbase = (wave >> 1) * 16;
  const int cb = (wave & 1) * 4;

  for (int j = 0; j < NCHUNK; ++j) {
    // store s_j (chunk-j start state)
    float* sdst = wsS + (size_t)j * (BATCHN * NXD);
    for (int q = tid; q < (BATCHN * NXD) / 4; q += 256) {
      int b = q >> 5, i = (q & 31) << 2;
      float4 v = *(const float4*)&sX[b * STX + i];
      *(float4*)&sdst[b * NXD + i] = v;
    }
    // S = S @ (A^32)^T + c_j
    v8f acc[4];
#pragma unroll
    for (int jj = 0; jj < 4; ++jj) { v8f z = {}; acc[jj] = z; }
    for (int kk = 0; kk < NXD; kk += 4) {
      v2f a = ldA(sX, STX, mbase, kk, lane);
#pragma unroll
      for (int jj = 0; jj < 4; ++jj)
        acc[jj] = wmma4(a, ldB(sAT, STB, kk, (cb + jj) * 16, lane), acc[jj]);
    }
    const float* cj = wsC + (size_t)j * (BATCHN * NXD);
#pragma unroll
    for (int jj = 0; jj < 4; ++jj) {
      int n = (cb + jj) * 16 + (lane & 15);
      int mo = mbase + ((lane >> 4) << 3);
#pragma unroll
      for (int r = 0; r < 8; ++r) acc[jj][r] += cj[(mo + r) * NXD + n];
    }
    __syncthreads();
#pragma unroll
    for (int jj = 0; jj < 4; ++jj) stAccLds(sX, STX, mbase, (cb + jj) * 16, lane, acc[jj]);
    __syncthreads();
  }
}

// ---------------------------------------------------------------------------
// Kernel 4: replay chunk j from s_j; emit y_t, x_t (and x_N from last chunk)
// ---------------------------------------------------------------------------
__global__ __launch_bounds__(256) void lti_phase2_kernel(const float* __restrict__ d,
                                                         const float* __restrict__ A,
                                                         const float* __restrict__ B,
                                                         const float* __restrict__ C,
                                                         const float* __restrict__ Dm,
                                                         const float* __restrict__ wsS,
                                                         float* __restrict__ out) {
  __shared__ float sAT[NXD * STB];
  __shared__ float sBT[NUD * STB];
  __shared__ float sCT[NXD * NYD];   // stride 32: half-waves hit disjoint bank halves
  __shared__ float sDT[NUD * NYD];
  __shared__ float sX[BATCHN * STX];
  const int tid = threadIdx.x, lane = tid & 31, wave = tid >> 5;
  const int chunk = blockIdx.x;

  for (int i = tid; i < NXD * NXD; i += 256) {
    int n = i >> 7, k = i & (NXD - 1);
    sAT[k * STB + n] = A[i];
  }
  for (int i = tid; i < NXD * NUD; i += 256) {
    int n = i >> 5, k = i & (NUD - 1);
    sBT[k * STB + n] = B[i];
  }
  for (int i = tid; i < NYD * NXD; i += 256) {
    int n = i >> 7, k = i & (NXD - 1);
    sCT[k * NYD + n] = C[i];
  }
  for (int i = tid; i < NYD * NUD; i += 256) {
    int n = i >> 5, k = i & (NUD - 1);
    sDT[k * NYD + n] = Dm[i];
  }
  {
    const float* ssrc = wsS + (size_t)chunk * (BATCHN * NXD);
    for (int q = tid; q < (BATCHN * NXD) / 4; q += 256) {
      int b = q >> 5, i = (q & 31) << 2;
      *(float4*)&sX[b * STX + i] = *(const float4*)&ssrc[b * NXD + i];
    }
  }
  __syncthreads();

  float* yout = out;                                 // [BATCH][NT][NY]
  float* xout = out + (size_t)BATCHN * NT * NYD;     // [BATCH][NT+1][NX]

  const int mbase = (wave >> 1) * 16;
  const int cb = (wave & 1) * 4;      // X column tiles
  const int ybase = (wave & 1) * 16;  // Y column tile
  const int t0 = chunk * CLEN;

  for (int s = 0; s < CLEN; ++s) {
    const int t = t0 + s;
    // emit x_t (pre-update state)
    for (int q = tid; q < (BATCHN * NXD) / 4; q += 256) {
      int b = q >> 5, i = (q & 31) << 2;
      float4 v = *(const float4*)&sX[b * STX + i];
      *(float4*)&xout[(size_t)b * (NT + 1) * NXD + (size_t)t * NXD + i] = v;
    }

    v8f acc[4], accY;
#pragma unroll
    for (int j = 0; j < 4; ++j) { v8f z = {}; acc[j] = z; }
    { v8f z = {}; accY = z; }

    // X@A^T and X@C^T share the A-operand
    for (int kk = 0; kk < NXD; kk += 4) {
      v2f a = ldA(sX, STX, mbase, kk, lane);
#pragma unroll
      for (int j = 0; j < 4; ++j)
        acc[j] = wmma4(a, ldB(sAT, STB, kk, (cb + j) * 16, lane), acc[j]);
      accY = wmma4(a, ldB(sCT, NYD, kk, ybase, lane), accY);
    }
    // U@B^T and U@D^T share the U operand
    for (int kk = 0; kk < NUD; kk += 4) {
      int m = mbase + (lane & 15);
      int k = kk + ((lane >> 4) << 1);
      const float* p = d + ((size_t)m * NT + t) * NUD + k;
      v2f a; a.x = p[0]; a.y = p[1];
#pragma unroll
      for (int j = 0; j < 4; ++j)
        acc[j] = wmma4(a, ldB(sBT, STB, kk, (cb + j) * 16, lane), acc[j]);
      accY = wmma4(a, ldB(sDT, NYD, kk, ybase, lane), accY);
    }

    // emit y_t
    {
      int jn = ybase + (lane & 15);
      int bo = mbase + ((lane >> 4) << 3);
#pragma unroll
      for (int r = 0; r < 8; ++r)
        yout[(size_t)(bo + r) * NT * NYD + (size_t)t * NYD + jn] = accY[r];
    }

    __syncthreads();
#pragma unroll
    for (int j = 0; j < 4; ++j) stAccLds(sX, STX, mbase, (cb + j) * 16, lane, acc[j]);
    __syncthreads();
  }

  // last chunk emits x_N (post-update final state)
  if (chunk == NCHUNK - 1) {
    for (int q = tid; q < (BATCHN * NXD) / 4; q += 256) {
      int b = q >> 5, i = (q & 31) << 2;
      float4 v = *(const float4*)&sX[b * STX + i];
      *(float4*)&xout[(size_t)b * (NT + 1) * NXD + (size_t)NT * NXD + i] = v;
    }
  }
}

// ---------------------------------------------------------------------------
extern "C" void kernel_launch(void* const* d_in, const int* in_sizes, int n_in,
                              void* d_out, int out_size, void* d_ws, size_t ws_size,
                              hipStream_t stream) {
  const float* d  = (const float*)d_in[0];
  const float* A  = (const float*)d_in[1];
  const float* B  = (const float*)d_in[2];
  const float* C  = (const float*)d_in[3];
  const float* Dm = (const float*)d_in[4];
  float* out = (float*)d_out;
  float* ws  = (float*)d_ws;

  float* wsApow = ws;                                   // 128*128 floats
  float* wsC    = wsApow + NXD * NXD;                   // NCHUNK*64*128 floats
  float* wsS    = wsC + (size_t)NCHUNK * BATCHN * NXD;  // NCHUNK*64*128 floats

  lti_power_kernel<<<1, 256, 0, stream>>>(A, wsApow);
  lti_phase1_kernel<<<NCHUNK, 256, 0, stream>>>(d, A, B, wsC);
  lti_combine_kernel<<<1, 256, 0, stream>>>(wsApow, wsC, wsS);
  lti_phase2_kernel<<<NCHUNK, 256, 0, stream>>>(d, A, B, C, Dm, wsS, out);
}